// DeformerNetMP_44169443672116
// MI455X (gfx1250) — compile-verified
//
#include <hip/hip_runtime.h>
#include <math.h>

#define BATCH 16

typedef float v2f __attribute__((ext_vector_type(2)));
typedef float v8f __attribute__((ext_vector_type(8)));

__device__ __forceinline__ v8f wmma4(v2f a, v2f b, v8f c) {
  // D = A(16x4,f32) * B(4x16,f32) + C(16x16,f32)  -> v_wmma_f32_16x16x4_f32
  return __builtin_amdgcn_wmma_f32_16x16x4_f32(false, a, false, b, (short)0, c, false, false);
}

static inline unsigned cdiv(size_t a, size_t b) { return (unsigned)((a + b - 1) / b); }

// ---------------------------------------------------------------------------
// Elementwise / layout kernels
// ---------------------------------------------------------------------------

// (B, Ctot, N) -> (B, N, Cuse), taking the first Cuse channels
__global__ void transpose_kernel(const float* __restrict__ in, float* __restrict__ out,
                                 int Npts, int Ctot, int Cuse) {
  int i = blockIdx.x * blockDim.x + threadIdx.x;
  int total = BATCH * Npts * Cuse;
  if (i >= total) return;
  int c = i % Cuse;
  int n = (i / Cuse) % Npts;
  int b = i / (Cuse * Npts);
  out[i] = in[((size_t)b * Ctot + c) * (size_t)Npts + n];
}

__global__ void norms_kernel(const float* __restrict__ xyz, float* __restrict__ norms,
                             int total) {
  int i = blockIdx.x * blockDim.x + threadIdx.x;
  if (i >= total) return;
  const float* p = xyz + (size_t)i * 3;
  norms[i] = p[0] * p[0] + p[1] * p[1] + p[2] * p[2];
}

// ---------------------------------------------------------------------------
// Density: inv_d[b,i] = 2.5*bw*N / sum_j exp(-||xi-xj||^2/(2 bw^2))
// WMMA computes 16x16 dot-product tiles of the Gram matrix; row sums are
// accumulated in-register, so the BxNxN matrix is never materialized.
// ---------------------------------------------------------------------------
__global__ void density_kernel(const float* __restrict__ xyz, const float* __restrict__ norms,
                               float* __restrict__ inv_d, int Npts, float bw) {
  int gid = blockIdx.x * blockDim.x + threadIdx.x;
  int wave = gid >> 5, lane = gid & 31;
  int tiles = Npts >> 4;
  int b = wave / tiles;
  if (b >= BATCH) return;                 // wave-uniform
  int ti = (wave % tiles) * 16;
  const float* X = xyz + (size_t)b * Npts * 3;
  const float* Nb = norms + (size_t)b * Npts;
  int half = lane >> 4, l16 = lane & 15;
  int ka = half ? 2 : 0, kb = half ? 3 : 1;
  v2f a;
  {
    int m = ti + l16;
    a.x = (ka < 3) ? X[m * 3 + ka] : 0.f;
    a.y = (kb < 3) ? X[m * 3 + kb] : 0.f;
  }
  float ni[8];
#pragma unroll
  for (int r = 0; r < 8; ++r) ni[r] = Nb[ti + r + half * 8];
  float sumr[8] = {0.f, 0.f, 0.f, 0.f, 0.f, 0.f, 0.f, 0.f};
  float inv2 = 1.f / (2.f * bw * bw);
  for (int tj = 0; tj < tiles; ++tj) {
    int n = tj * 16 + l16;
    v2f bf;
    bf.x = (ka < 3) ? X[n * 3 + ka] : 0.f;
    bf.y = (kb < 3) ? X[n * 3 + kb] : 0.f;
    v8f dot = {};
    dot = wmma4(a, bf, dot);
    float nj = Nb[n];
#pragma unroll
    for (int r = 0; r < 8; ++r) {
      float d = ni[r] + nj - 2.f * dot[r];
      sumr[r] += __expf(-d * inv2);
    }
  }
#pragma unroll
  for (int r = 0; r < 8; ++r) {
    float v = sumr[r];
    for (int o = 1; o < 16; o <<= 1) v += __shfl_xor(v, o);
    sumr[r] = v;
  }
  if (l16 == 0) {
    float scale = 2.5f * bw * (float)Npts;
#pragma unroll
    for (int r = 0; r < 8; ++r)
      inv_d[(size_t)b * Npts + ti + half * 8 + r] = scale / sumr[r];
  }
}

// ---------------------------------------------------------------------------
// Farthest point sampling: one block per batch, sequential npoint iterations
// ---------------------------------------------------------------------------
__global__ void fps_kernel(const float* __restrict__ xyz, int* __restrict__ fidx,
                           int Npts, int npoint) {
  int b = blockIdx.x;
  const float* X = xyz + (size_t)b * Npts * 3;
  __shared__ float mind[2048];
  __shared__ float bv[256];
  __shared__ int bi[256];
  __shared__ int far_s;
  for (int j = threadIdx.x; j < Npts; j += blockDim.x) mind[j] = 1e10f;
  if (threadIdx.x == 0) far_s = 0;
  __syncthreads();
  for (int it = 0; it < npoint; ++it) {
    int far = far_s;
    if (threadIdx.x == 0) fidx[b * npoint + it] = far;
    float cx = X[far * 3 + 0], cy = X[far * 3 + 1], cz = X[far * 3 + 2];
    float best = -1.f;
    int bidx = 0x7fffffff;
    for (int j = threadIdx.x; j < Npts; j += blockDim.x) {
      float dx = X[j * 3 + 0] - cx, dy = X[j * 3 + 1] - cy, dz = X[j * 3 + 2] - cz;
      float d = fminf(mind[j], dx * dx + dy * dy + dz * dz);
      mind[j] = d;
      if (d > best) { best = d; bidx = j; }
    }
    bv[threadIdx.x] = best;
    bi[threadIdx.x] = bidx;
    __syncthreads();
    for (int o = 128; o > 0; o >>= 1) {
      if (threadIdx.x < (unsigned)o) {
        if (bv[threadIdx.x + o] > bv[threadIdx.x] ||
            (bv[threadIdx.x + o] == bv[threadIdx.x] && bi[threadIdx.x + o] < bi[threadIdx.x])) {
          bv[threadIdx.x] = bv[threadIdx.x + o];
          bi[threadIdx.x] = bi[threadIdx.x + o];
        }
      }
      __syncthreads();
    }
    if (threadIdx.x == 0) far_s = bi[0];
    __syncthreads();
  }
}

__global__ void gather_xyz_kernel(const float* __restrict__ xyz, const int* __restrict__ fidx,
                                  float* __restrict__ new_xyz, int Npts, int S) {
  int i = blockIdx.x * blockDim.x + threadIdx.x;
  if (i >= BATCH * S) return;
  int b = i / S;
  int j = fidx[i];
  const float* src = xyz + ((size_t)b * Npts + j) * 3;
  new_xyz[i * 3 + 0] = src[0];
  new_xyz[i * 3 + 1] = src[1];
  new_xyz[i * 3 + 2] = src[2];
}

// ---------------------------------------------------------------------------
// kNN (32/64 smallest distances): block per query, LDS distance cache +
// iterated argmin with exclusion (stable, lowest-index tie-break like top_k)
// ---------------------------------------------------------------------------
__global__ void knn_kernel(const float* __restrict__ new_xyz, const float* __restrict__ xyz,
                           int* __restrict__ knn, int S, int Npts, int nsample) {
  int b = blockIdx.x / S, s = blockIdx.x % S;
  __shared__ float dist[2048];
  __shared__ float bestv[256];
  __shared__ int besti[256];
  const float* q = new_xyz + ((size_t)b * S + s) * 3;
  float qx = q[0], qy = q[1], qz = q[2];
  const float* X = xyz + (size_t)b * Npts * 3;
  for (int j = threadIdx.x; j < Npts; j += blockDim.x) {
    float dx = X[j * 3 + 0] - qx, dy = X[j * 3 + 1] - qy, dz = X[j * 3 + 2] - qz;
    dist[j] = dx * dx + dy * dy + dz * dz;
  }
  __syncthreads();
  int* out = knn + ((size_t)b * S + s) * nsample;
  for (int t = 0; t < nsample; ++t) {
    float bvv = 3e38f;
    int bii = 0x7fffffff;
    for (int j = threadIdx.x; j < Npts; j += blockDim.x) {
      float v = dist[j];
      if (v < bvv) { bvv = v; bii = j; }
    }
    bestv[threadIdx.x] = bvv;
    besti[threadIdx.x] = bii;
    __syncthreads();
    for (int o = 128; o > 0; o >>= 1) {
      if (threadIdx.x < (unsigned)o) {
        if (bestv[threadIdx.x + o] < bestv[threadIdx.x] ||
            (bestv[threadIdx.x + o] == bestv[threadIdx.x] &&
             besti[threadIdx.x + o] < besti[threadIdx.x])) {
          bestv[threadIdx.x] = bestv[threadIdx.x + o];
          besti[threadIdx.x] = besti[threadIdx.x + o];
        }
      }
      __syncthreads();
    }
    if (threadIdx.x == 0) {
      out[t] = besti[0];
      dist[besti[0]] = 3e38f;
    }
    __syncthreads();
  }
}

// ---------------------------------------------------------------------------
// Grouping
// ---------------------------------------------------------------------------
__global__ void group_kernel(const float* __restrict__ xyz, const float* __restrict__ pts,
                             const float* __restrict__ inv_d, const float* __restrict__ new_xyz,
                             const int* __restrict__ knn, float* __restrict__ new_pts,
                             float* __restrict__ g_xyz, float* __restrict__ g_den,
                             int S, int K, int Npts, int C) {
  size_t i = blockIdx.x * (size_t)blockDim.x + threadIdx.x;
  size_t total = (size_t)BATCH * S * K;
  if (i >= total) return;
  int s = (int)((i / K) % S);
  int b = (int)(i / ((size_t)K * S));
  int j = knn[i];
  const float* q = new_xyz + ((size_t)b * S + s) * 3;
  const float* x = xyz + ((size_t)b * Npts + j) * 3;
  float gx = x[0] - q[0], gy = x[1] - q[1], gz = x[2] - q[2];
  g_xyz[i * 3 + 0] = gx;
  g_xyz[i * 3 + 1] = gy;
  g_xyz[i * 3 + 2] = gz;
  float* np = new_pts + i * (size_t)(3 + C);
  np[0] = gx; np[1] = gy; np[2] = gz;
  const float* p = pts + ((size_t)b * Npts + j) * C;
  for (int c = 0; c < C; ++c) np[3 + c] = p[c];
  g_den[i] = inv_d[(size_t)b * Npts + j];
}

__global__ void group_all_kernel(const float* __restrict__ xyz, const float* __restrict__ pts,
                                 const float* __restrict__ inv_d, float* __restrict__ new_pts,
                                 float* __restrict__ g_xyz, float* __restrict__ g_den,
                                 int Npts, int C) {
  size_t i = blockIdx.x * (size_t)blockDim.x + threadIdx.x;
  size_t total = (size_t)BATCH * Npts;
  if (i >= total) return;
  const float* x = xyz + i * 3;
  g_xyz[i * 3 + 0] = x[0];
  g_xyz[i * 3 + 1] = x[1];
  g_xyz[i * 3 + 2] = x[2];
  float* np = new_pts + i * (size_t)(3 + C);
  np[0] = x[0]; np[1] = x[1]; np[2] = x[2];
  const float* p = pts + i * (size_t)C;
  for (int c = 0; c < C; ++c) np[3 + c] = p[c];
  g_den[i] = inv_d[i];
}

// ---------------------------------------------------------------------------
// Generic fused GEMM + bias, fp32 WMMA 16x16x4.  One wave = one 16x16 C tile.
// Y(M,N) = A(M,K) @ W(K,N) + bias(N)
// Fast path for fully 16/8-aligned shapes (the lin layers and head FCs):
// no clamps, K unrolled x2, A-row prefetch into the WGP cache.
// ---------------------------------------------------------------------------
__global__ void gemm_bias_kernel(const float* __restrict__ A, const float* __restrict__ W,
                                 const float* __restrict__ bias, float* __restrict__ Y,
                                 int M, int K, int N) {
  int gid = blockIdx.x * blockDim.x + threadIdx.x;
  int wave = gid >> 5, lane = gid & 31;
  int ntn = (N + 15) >> 4;
  int tile_m = (wave / ntn) * 16;
  int tile_n = (wave % ntn) * 16;
  if (tile_m >= M) return;                 // wave-uniform
  int half = lane >> 4, l16 = lane & 15;
  int row = tile_m + l16;
  int col = tile_n + l16;
  v8f acc = {};
  if ((((M | N) & 15) == 0) && ((K & 7) == 0)) {
    const float* Arow = A + (size_t)row * K + (half ? 2 : 0);
    const float* Wcol = W + (size_t)(half ? 2 : 0) * N + col;
    for (int k = 0; k < K; k += 8) {
      if (k + 64 < K) __builtin_prefetch(Arow + k + 64, 0, 3);
      v2f a0, b0, a1, b1;
      a0.x = Arow[k + 0];
      a0.y = Arow[k + 1];
      a1.x = Arow[k + 4];
      a1.y = Arow[k + 5];
      b0.x = Wcol[(size_t)(k + 0) * N];
      b0.y = Wcol[(size_t)(k + 1) * N];
      b1.x = Wcol[(size_t)(k + 4) * N];
      b1.y = Wcol[(size_t)(k + 5) * N];
      acc = wmma4(a0, b0, acc);
      acc = wmma4(a1, b1, acc);
    }
  } else {
    for (int k = 0; k < K; k += 4) {
      int ka = k + (half ? 2 : 0);
      int kb = ka + 1;
      v2f af, bf;
      af.x = (row < M && ka < K) ? A[(size_t)row * K + ka] : 0.f;
      af.y = (row < M && kb < K) ? A[(size_t)row * K + kb] : 0.f;
      bf.x = (col < N && ka < K) ? W[(size_t)ka * N + col] : 0.f;
      bf.y = (col < N && kb < K) ? W[(size_t)kb * N + col] : 0.f;
      acc = wmma4(af, bf, acc);
    }
  }
  if (col >= N) return;
  float bvv = bias ? bias[col] : 0.f;
#pragma unroll
  for (int r = 0; r < 8; ++r) {
    int rr = tile_m + r + half * 8;
    if (rr < M) Y[(size_t)rr * N + col] = acc[r] + bvv;
  }
}

// per-channel mean/var over M rows (batch-norm over axes (0,1,2) / (0,1))
__global__ void colstats_kernel(const float* __restrict__ Y, float* __restrict__ mean,
                                float* __restrict__ var, int M, int N) {
  int c = blockIdx.x;
  float s = 0.f, s2 = 0.f;
  for (int r = threadIdx.x; r < M; r += blockDim.x) {
    float v = Y[(size_t)r * N + c];
    s += v;
    s2 += v * v;
  }
  __shared__ float ss[256];
  __shared__ float ss2[256];
  ss[threadIdx.x] = s;
  ss2[threadIdx.x] = s2;
  __syncthreads();
  for (int o = 128; o > 0; o >>= 1) {
    if (threadIdx.x < (unsigned)o) {
      ss[threadIdx.x] += ss[threadIdx.x + o];
      ss2[threadIdx.x] += ss2[threadIdx.x + o];
    }
    __syncthreads();
  }
  if (threadIdx.x == 0) {
    float m = ss[0] / (float)M;
    mean[c] = m;
    var[c] = ss2[0] / (float)M - m * m;
  }
}

__global__ void bnrelu_kernel(float* __restrict__ Y, const float* __restrict__ mean,
                              const float* __restrict__ var, const float* __restrict__ g,
                              const float* __restrict__ beta, int M, int N) {
  size_t i = blockIdx.x * (size_t)blockDim.x + threadIdx.x;
  if (i >= (size_t)M * N) return;
  int c = (int)(i % N);
  float v = g[c] * (Y[i] - mean[c]) * rsqrtf(var[c] + 1e-5f) + beta[c];
  Y[i] = fmaxf(v, 0.f);
}

// ds = g_den / max_k g_den : one wave per (b,s) group
__global__ void dsinit_kernel(const float* __restrict__ gden, float* __restrict__ ds,
                              int groups, int K) {
  int gid = blockIdx.x * blockDim.x + threadIdx.x;
  int wave = gid >> 5, lane = gid & 31;
  if (wave >= groups) return;
  const float* g = gden + (size_t)wave * K;
  float mx = -3e38f;
  for (int k = lane; k < K; k += 32) mx = fmaxf(mx, g[k]);
  for (int o = 16; o > 0; o >>= 1) mx = fmaxf(mx, __shfl_xor(mx, o));
  for (int k = lane; k < K; k += 32) ds[(size_t)wave * K + k] = g[k] / mx;
}

__global__ void mul_bcast_kernel(float* __restrict__ h, const float* __restrict__ ds,
                                 size_t M, int C) {
  size_t i = blockIdx.x * (size_t)blockDim.x + threadIdx.x;
  if (i >= M * (size_t)C) return;
  h[i] *= ds[i / C];
}

// agg[bs, c, w] = sum_k h[bs,k,c] * wts[bs,k,w]; K in {32,64,128} -> unroll x2
__global__ void einsum_kernel(const float* __restrict__ h, const float* __restrict__ w,
                              float* __restrict__ agg, int BS, int K, int Cout) {
  int gid = blockIdx.x * blockDim.x + threadIdx.x;
  int wave = gid >> 5, lane = gid & 31;
  int ctiles = Cout >> 4;
  int bs = wave / ctiles;
  if (bs >= BS) return;                    // wave-uniform
  int tile_c = (wave % ctiles) * 16;
  int half = lane >> 4, l16 = lane & 15;
  const float* H = h + (size_t)bs * K * Cout + tile_c + l16;
  const float* Wt = w + (size_t)bs * K * 16 + l16;
  int kofs = half ? 2 : 0;
  v8f acc = {};
  for (int k = 0; k < K; k += 8) {
    int k0 = k + kofs;
    int k1 = k0 + 4;
    v2f a0, b0, a1, b1;
    a0.x = H[(size_t)(k0 + 0) * Cout];
    a0.y = H[(size_t)(k0 + 1) * Cout];
    a1.x = H[(size_t)(k1 + 0) * Cout];
    a1.y = H[(size_t)(k1 + 1) * Cout];
    b0.x = Wt[(k0 + 0) * 16];
    b0.y = Wt[(k0 + 1) * 16];
    b1.x = Wt[(k1 + 0) * 16];
    b1.y = Wt[(k1 + 1) * 16];
    acc = wmma4(a0, b0, acc);
    acc = wmma4(a1, b1, acc);
  }
  float* O = agg + (size_t)bs * Cout * 16;
#pragma unroll
  for (int r = 0; r < 8; ++r) {
    int c = tile_c + r + half * 8;
    O[c * 16 + l16] = acc[r];
  }
}

// per-row (last-axis) group norm + relu: one wave per row
__global__ void gnorm_relu_kernel(float* __restrict__ Y, const float* __restrict__ g,
                                  const float* __restrict__ bt, int M, int C) {
  int gid = blockIdx.x * blockDim.x + threadIdx.x;
  int wave = gid >> 5, lane = gid & 31;
  if (wave >= M) return;
  float* row = Y + (size_t)wave * C;
  float s = 0.f, s2 = 0.f;
  for (int c = lane; c < C; c += 32) {
    float v = row[c];
    s += v;
    s2 += v * v;
  }
  for (int o = 16; o > 0; o >>= 1) {
    s += __shfl_xor(s, o);
    s2 += __shfl_xor(s2, o);
  }
  float m = s / (float)C;
  float inv = rsqrtf(s2 / (float)C - m * m + 1e-5f);
  for (int c = lane; c < C; c += 32) {
    float v = g[c] * (row[c] - m) * inv + bt[c];
    row[c] = fmaxf(v, 0.f);
  }
}

__global__ void concat_head_kernel(const float* __restrict__ x, const float* __restrict__ g,
                                   const float* __restrict__ gr, float* __restrict__ out) {
  int i = blockIdx.x * blockDim.x + threadIdx.x;
  if (i >= 16 * 544) return;
  int b = i / 544, c = i % 544;
  float v;
  if (c < 256) v = x[b * 256 + c];
  else if (c < 512) v = g[b * 256 + (c - 256)];
  else v = gr[b * 32 + (c - 512)];
  out[i] = v;
}

// translation + rot6d
__global__ void head_out_kernel(const float* __restrict__ h9, float* __restrict__ out) {
  int b = blockIdx.x * blockDim.x + threadIdx.x;
  if (b >= 16) return;
  const float* o = h9 + b * 9;
  out[b * 3 + 0] = o[0];
  out[b * 3 + 1] = o[1];
  out[b * 3 + 2] = o[2];
  float xv[3] = {o[0], o[1], o[2]};
  float n = fmaxf(sqrtf(xv[0] * xv[0] + xv[1] * xv[1] + xv[2] * xv[2]), 1e-8f);
  xv[0] /= n; xv[1] /= n; xv[2] /= n;
  float a[3] = {o[3], o[4], o[5]};
  float z[3] = {xv[1] * a[2] - xv[2] * a[1], xv[2] * a[0] - xv[0] * a[2],
                xv[0] * a[1] - xv[1] * a[0]};
  float nz = fmaxf(sqrtf(z[0] * z[0] + z[1] * z[1] + z[2] * z[2]), 1e-8f);
  z[0] /= nz; z[1] /= nz; z[2] /= nz;
  float y[3] = {z[1] * xv[2] - z[2] * xv[1], z[2] * xv[0] - z[0] * xv[2],
                z[0] * xv[1] - z[1] * xv[0]};
  float* R = out + 48 + b * 9;
  R[0] = xv[0]; R[1] = y[0]; R[2] = z[0];
  R[3] = xv[1]; R[4] = y[1]; R[5] = z[1];
  R[6] = xv[2]; R[7] = y[2]; R[8] = z[2];
}

// ---------------------------------------------------------------------------
// Host-side orchestration
// ---------------------------------------------------------------------------

struct SaP {
  const float *mlp_w, *mlp_b, *mlp_g, *mlp_bt;
  const float *wn_w[3], *wn_b[3], *wn_g[3], *wn_bt[3];
  const float *dn_w[3], *dn_b[3], *dn_g[3], *dn_bt[3];
  const float *lin_w, *lin_b, *lbn_g, *lbn_b;
};

// jax pytree leaf order for the SA dict (keys sorted):
// dn(12), lbn_b, lbn_g, lin_b, lin_w, mlp_b, mlp_beta, mlp_g, mlp_w, wn(12)
static SaP load_sa(void* const* din, int base) {
  SaP p;
  int i = base;
  for (int l = 0; l < 3; ++l) {
    p.dn_w[l] = (const float*)din[i++]; p.dn_b[l] = (const float*)din[i++];
    p.dn_g[l] = (const float*)din[i++]; p.dn_bt[l] = (const float*)din[i++];
  }
  p.lbn_b = (const float*)din[i++]; p.lbn_g = (const float*)din[i++];
  p.lin_b = (const float*)din[i++]; p.lin_w = (const float*)din[i++];
  p.mlp_b = (const float*)din[i++]; p.mlp_bt = (const float*)din[i++];
  p.mlp_g = (const float*)din[i++]; p.mlp_w = (const float*)din[i++];
  for (int l = 0; l < 3; ++l) {
    p.wn_w[l] = (const float*)din[i++]; p.wn_b[l] = (const float*)din[i++];
    p.wn_g[l] = (const float*)din[i++]; p.wn_bt[l] = (const float*)din[i++];
  }
  return p;
}

struct Bufs {
  float *xyz0, *pts0, *xyz1, *f1, *xyz2, *f2;
  float *norms, *invd;
  int *fidx, *knn;
  float *gxyz, *newpts, *gden, *h, *dsA, *dsB, *wtsA, *wtsB, *agg;
  float *mean, *var;
};

static void gemm(const float* A, const float* W, const float* bias, float* Y,
                 size_t M, int K, int N, hipStream_t s) {
  size_t tm = (M + 15) / 16, tn = ((size_t)N + 15) / 16;
  size_t waves = tm * tn;
  gemm_bias_kernel<<<cdiv(waves * 32, 128), 128, 0, s>>>(A, W, bias, Y, (int)M, K, N);
}

static void bnr(float* Y, const float* g, const float* bt, size_t M, int N,
                Bufs& B_, hipStream_t s) {
  colstats_kernel<<<N, 256, 0, s>>>(Y, B_.mean, B_.var, (int)M, N);
  bnrelu_kernel<<<cdiv(M * (size_t)N, 256), 256, 0, s>>>(Y, B_.mean, B_.var, g, bt, (int)M, N);
}

static void sa_layer(const SaP& p, const float* xyz, const float* pts, int Npts, int C,
                     int S, int K, int cout, float bw, bool gall, float* new_xyz,
                     float* outf, Bufs& W, hipStream_t s) {
  int cin = 3 + C;
  size_t Mrows = (size_t)BATCH * S * K;
  // density -> inv_d
  norms_kernel<<<cdiv((size_t)BATCH * Npts, 256), 256, 0, s>>>(xyz, W.norms, BATCH * Npts);
  size_t dwaves = (size_t)BATCH * (Npts / 16);
  density_kernel<<<cdiv(dwaves * 32, 128), 128, 0, s>>>(xyz, W.norms, W.invd, Npts, bw);
  if (!gall) {
    fps_kernel<<<BATCH, 256, 0, s>>>(xyz, W.fidx, Npts, S);
    gather_xyz_kernel<<<cdiv((size_t)BATCH * S, 256), 256, 0, s>>>(xyz, W.fidx, new_xyz, Npts, S);
    knn_kernel<<<BATCH * S, 256, 0, s>>>(new_xyz, xyz, W.knn, S, Npts, K);
    group_kernel<<<cdiv(Mrows, 256), 256, 0, s>>>(xyz, pts, W.invd, new_xyz, W.knn,
                                                  W.newpts, W.gxyz, W.gden, S, K, Npts, C);
  } else {
    group_all_kernel<<<cdiv(Mrows, 256), 256, 0, s>>>(xyz, pts, W.invd, W.newpts, W.gxyz,
                                                      W.gden, Npts, C);
  }
  // point MLP + batch-norm(0,1,2) + relu
  gemm(W.newpts, p.mlp_w, p.mlp_b, W.h, Mrows, cin, cout, s);
  bnr(W.h, p.mlp_g, p.mlp_bt, Mrows, cout, W, s);
  // density net: 1 -> 16 -> 8 -> 1
  dsinit_kernel<<<cdiv((size_t)BATCH * S * 32, 256), 256, 0, s>>>(W.gden, W.dsA, BATCH * S, K);
  const int dsc[4] = {1, 16, 8, 1};
  float* din = W.dsA;
  float* dout = W.dsB;
  for (int l = 0; l < 3; ++l) {
    gemm(din, p.dn_w[l], p.dn_b[l], dout, Mrows, dsc[l], dsc[l + 1], s);
    bnr(dout, p.dn_g[l], p.dn_bt[l], Mrows, dsc[l + 1], W, s);
    float* t = din; din = dout; dout = t;
  }
  mul_bcast_kernel<<<cdiv(Mrows * (size_t)cout, 256), 256, 0, s>>>(W.h, din, Mrows, cout);
  // weight net: 3 -> 8 -> 8 -> 16
  gemm(W.gxyz, p.wn_w[0], p.wn_b[0], W.wtsA, Mrows, 3, 8, s);
  bnr(W.wtsA, p.wn_g[0], p.wn_bt[0], Mrows, 8, W, s);
  gemm(W.wtsA, p.wn_w[1], p.wn_b[1], W.wtsB, Mrows, 8, 8, s);
  bnr(W.wtsB, p.wn_g[1], p.wn_bt[1], Mrows, 8, W, s);
  gemm(W.wtsB, p.wn_w[2], p.wn_b[2], W.wtsA, Mrows, 8, 16, s);
  bnr(W.wtsA, p.wn_g[2], p.wn_bt[2], Mrows, 16, W, s);
  // aggregation einsum + linear + batch-norm(0,1) + relu
  size_t ewaves = (size_t)BATCH * S * (cout / 16);
  einsum_kernel<<<cdiv(ewaves * 32, 128), 128, 0, s>>>(W.h, W.wtsA, W.agg, BATCH * S, K, cout);
  gemm(W.agg, p.lin_w, p.lin_b, outf, (size_t)BATCH * S, 16 * cout, cout, s);
  bnr(outf, p.lbn_g, p.lbn_b, (size_t)BATCH * S, cout, W, s);
}

static void run_branch(const float* pc, int C0, const SaP& p1, const SaP& p2, const SaP& p3,
                       float* outfeat, Bufs& B_, hipStream_t s) {
  transpose_kernel<<<cdiv((size_t)BATCH * 2048 * 3, 256), 256, 0, s>>>(pc, B_.xyz0, 2048, C0, 3);
  transpose_kernel<<<cdiv((size_t)BATCH * 2048 * C0, 256), 256, 0, s>>>(pc, B_.pts0, 2048, C0, C0);
  sa_layer(p1, B_.xyz0, B_.pts0, 2048, C0, 512, 32, 64, 0.1f, false, B_.xyz1, B_.f1, B_, s);
  sa_layer(p2, B_.xyz1, B_.f1, 512, 64, 128, 64, 128, 0.2f, false, B_.xyz2, B_.f2, B_, s);
  sa_layer(p3, B_.xyz2, B_.f2, 128, 128, 1, 128, 256, 0.4f, true, nullptr, outfeat, B_, s);
}

extern "C" void kernel_launch(void* const* d_in, const int* in_sizes, int n_in,
                              void* d_out, int out_size, void* d_ws, size_t ws_size,
                              hipStream_t stream) {
  (void)in_sizes; (void)n_in; (void)out_size; (void)ws_size;
  char* base = (char*)d_ws;
  size_t off = 0;
  auto alloc = [&](size_t nfloats) -> float* {
    float* r = (float*)(base + off);
    off += nfloats * sizeof(float);
    off = (off + 255) & ~(size_t)255;
    return r;
  };

  Bufs B_;
  B_.xyz0 = alloc(98304);        // 16*2048*3
  B_.pts0 = alloc(131072);       // 16*2048*4
  B_.xyz1 = alloc(24576);        // 16*512*3
  B_.f1 = alloc(524288);         // 16*512*64
  B_.xyz2 = alloc(6144);         // 16*128*3
  B_.f2 = alloc(262144);         // 16*128*128
  B_.norms = alloc(32768);
  B_.invd = alloc(32768);
  B_.fidx = (int*)alloc(8192);
  B_.knn = (int*)alloc(262144);
  B_.gxyz = alloc(786432);
  B_.newpts = alloc(8781824);    // max: 16*128*64*67
  B_.gden = alloc(262144);
  B_.h = alloc(16777216);        // max: 16*512*32*64
  B_.dsA = alloc(4194304);
  B_.dsB = alloc(4194304);
  B_.wtsA = alloc(4194304);
  B_.wtsB = alloc(4194304);
  B_.agg = alloc(8388608);       // max: 16*512*1024
  B_.mean = alloc(1024);
  B_.var = alloc(1024);
  float* xfeat = alloc(4096);    // 16*256
  float* gfeat = alloc(4096);
  float* grb = alloc(512);       // 16*32
  float* grb2 = alloc(512);
  float* hcat = alloc(16 * 544);
  float* t1 = alloc(16 * 256);
  float* t2 = alloc(16 * 128);
  float* t3 = alloc(16 * 64);
  float* h9 = alloc(256);

  // input pointer mapping (top-level insertion order; nested = pytree order)
  SaP sa1 = load_sa(d_in, 3);
  SaP sa2 = load_sa(d_in, 31);
  SaP sa3 = load_sa(d_in, 59);
  SaP sa1g = load_sa(d_in, 87);
  SaP sa2g = load_sa(d_in, 115);
  SaP sa3g = load_sa(d_in, 143);
  const int HB = 171;
  const float* fc1_w = (const float*)d_in[HB + 0];
  const float* fc1_b = (const float*)d_in[HB + 1];
  const float* fc1g_w = (const float*)d_in[HB + 2];
  const float* fc1g_b = (const float*)d_in[HB + 3];
  const float* fc2g_w = (const float*)d_in[HB + 4];
  const float* fc2g_b = (const float*)d_in[HB + 5];
  const float* fc3_w = (const float*)d_in[HB + 6];
  const float* fc3_b = (const float*)d_in[HB + 7];
  const float* fc4_w = (const float*)d_in[HB + 8];
  const float* fc4_b = (const float*)d_in[HB + 9];
  const float* fc5_w = (const float*)d_in[HB + 10];
  const float* fc5_b = (const float*)d_in[HB + 11];
  const float* gn1_g = (const float*)d_in[HB + 12];
  const float* gn1_b = (const float*)d_in[HB + 13];
  const float* gn1g_g = (const float*)d_in[HB + 14];
  const float* gn1g_b = (const float*)d_in[HB + 15];
  const float* gn2g_g = (const float*)d_in[HB + 16];
  const float* gn2g_b = (const float*)d_in[HB + 17];
  const float* gn3_g = (const float*)d_in[HB + 18];
  const float* gn3_b = (const float*)d_in[HB + 19];
  const float* gn4_g = (const float*)d_in[HB + 20];
  const float* gn4_b = (const float*)d_in[HB + 21];

  // two branches
  run_branch((const float*)d_in[0], 4, sa1, sa2, sa3, xfeat, B_, stream);
  run_branch((const float*)d_in[1], 3, sa1g, sa2g, sa3g, gfeat, B_, stream);

  // head
  const float* grip = (const float*)d_in[2];
  gemm(grip, fc1g_w, fc1g_b, grb, 16, 3, 32, stream);
  gnorm_relu_kernel<<<2, 256, 0, stream>>>(grb, gn1g_g, gn1g_b, 16, 32);
  gemm(grb, fc2g_w, fc2g_b, grb2, 16, 32, 32, stream);
  gnorm_relu_kernel<<<2, 256, 0, stream>>>(grb2, gn2g_g, gn2g_b, 16, 32);
  concat_head_kernel<<<cdiv(16 * 544, 256), 256, 0, stream>>>(xfeat, gfeat, grb2, hcat);
  gemm(hcat, fc1_w, fc1_b, t1, 16, 544, 256, stream);
  gnorm_relu_kernel<<<2, 256, 0, stream>>>(t1, gn1_g, gn1_b, 16, 256);
  gemm(t1, fc3_w, fc3_b, t2, 16, 256, 128, stream);
  gnorm_relu_kernel<<<2, 256, 0, stream>>>(t2, gn3_g, gn3_b, 16, 128);
  gemm(t2, fc4_w, fc4_b, t3, 16, 128, 64, stream);
  gnorm_relu_kernel<<<2, 256, 0, stream>>>(t3, gn4_g, gn4_b, 16, 64);
  gemm(t3, fc5_w, fc5_b, h9, 16, 64, 9, stream);
  head_out_kernel<<<1, 32, 0, stream>>>(h9, (float*)d_out);
}